// DeadRecsGNN_62938450755871
// MI455X (gfx1250) — compile-verified
//
#include <hip/hip_runtime.h>

// ---------------------------------------------------------------------------
// DeadRecs hetero-GraphSAGE (2 layers, 4 edge types, D=128, f32)
// Phase 1 per conv: edge scatter-add (segment mean) -> agg/cnt   [HBM/L2 bound]
// Phase 2 per conv: h += (agg/cnt) @ Wl^T + b + x_dst @ Wr^T via
//                   V_WMMA_F32_16X16X4_F32 (exact f32, wave32),
//                   64-row M-blocking: 1 weight fragment feeds 4 WMMAs.
// ---------------------------------------------------------------------------

typedef __attribute__((ext_vector_type(2))) float v2f;
typedef __attribute__((ext_vector_type(8))) float v8f;

namespace {
constexpr int D = 128;
constexpr int N_SHOW = 200000;
constexpr int N_SONG = 100000;
constexpr int N_VENUE = 20000;
constexpr int MT = 64;        // rows of dst nodes per block (4 WMMA sub-tiles)
constexpr int LDP = D + 4;    // LDS row pitch: 132 floats -> conflict-free, 16B aligned
}

__global__ void zero_f32(float* __restrict__ p, long long n) {
  long long i = (long long)blockIdx.x * blockDim.x + threadIdx.x;
  long long stride = (long long)gridDim.x * blockDim.x;
  for (; i < n; i += stride) p[i] = 0.0f;
}

__global__ void relu_f32(float* __restrict__ p, long long n) {
  long long i = (long long)blockIdx.x * blockDim.x + threadIdx.x;
  long long stride = (long long)gridDim.x * blockDim.x;
  for (; i < n; i += stride) p[i] = fmaxf(p[i], 0.0f);
}

// One wave (32 lanes) per edge: lane handles 4 consecutive features.
// Coalesced 512B gather of the src row, atomic scatter-add into agg[dst].
__global__ __launch_bounds__(256)
void scatter_mean_accum(const float* __restrict__ xsrc,
                        const int* __restrict__ src,
                        const int* __restrict__ dst,
                        float* __restrict__ agg,
                        float* __restrict__ cnt,
                        int E) {
  long long gid = (long long)blockIdx.x * blockDim.x + threadIdx.x;
  int edge = (int)(gid >> 5);
  int lane = (int)(gid & 31);
  if (edge >= E) return;
  int s = src[edge];
  int d = dst[edge];
  const float4* x4 = reinterpret_cast<const float4*>(xsrc);
  float4 v = x4[(long long)s * (D / 4) + lane];
  float* ap = agg + (long long)d * D + lane * 4;
  atomicAdd(ap + 0, v.x);
  atomicAdd(ap + 1, v.y);
  atomicAdd(ap + 2, v.z);
  atomicAdd(ap + 3, v.w);
  if (lane == 0) atomicAdd(cnt + d, 1.0f);
}

// Block = 256 threads = 8 wave32s. Block owns a 64-row tile of dst nodes
// (4 sub-tiles of 16); wave w owns output columns [16w, 16w+16). Each k-step
// loads one B (weight) fragment and issues 4 chained WMMAs, one per sub-tile:
//   h[tile] += mean @ Wl^T + b + x_dst @ Wr^T
__global__ __launch_bounds__(256)
void sage_wmma_gemm(const float* __restrict__ agg,
                    const float* __restrict__ cnt,
                    const float* __restrict__ xdst,
                    const float* __restrict__ Wl,
                    const float* __restrict__ bias,
                    const float* __restrict__ Wr,
                    float* __restrict__ h,
                    int n_dst) {
  // Pitch 132: lane r reads row base bank (132*r + k) % 64 = 4r + c -> all 16
  // rows distinct; the two K-halves interleave onto the remaining banks.
  __shared__ float Am[MT][LDP];  // mean-aggregate tile (pre-division)
  __shared__ float Ar[MT][LDP];  // root (x_dst) tile

  const int tid = threadIdx.x;
  const int row0 = blockIdx.x * MT;

  // Cooperative fill: float4 gather -> b128 LDS stores. Clamped rows keep
  // EXEC all-ones for the WMMA region (out-of-range rows never stored).
  {
    const float4* a4 = reinterpret_cast<const float4*>(agg);
    const float4* x4 = reinterpret_cast<const float4*>(xdst);
    for (int t = tid; t < MT * (D / 4); t += 256) {
      int r = t >> 5;            // row within tile
      int k4 = t & 31;           // float4 index within row
      int gr = min(row0 + r, n_dst - 1);
      *reinterpret_cast<float4*>(&Am[r][k4 * 4]) = a4[(long long)gr * (D / 4) + k4];
      *reinterpret_cast<float4*>(&Ar[r][k4 * 4]) = x4[(long long)gr * (D / 4) + k4];
    }
  }
  __syncthreads();

  const int wave = tid >> 5;
  const int lane = tid & 31;
  const int half = lane >> 4;   // selects K pair within the 16x4 fragment
  const int r = lane & 15;      // A row within sub-tile == C/D column index
  const int n = wave * 16 + r;  // global output column (B column)

  float invc[4];
  #pragma unroll
  for (int mt = 0; mt < 4; ++mt) {
    float c = cnt[min(row0 + mt * 16 + r, n_dst - 1)];
    invc[mt] = 1.0f / fmaxf(c, 1.0f);
  }

  v8f acc[4] = {v8f{}, v8f{}, v8f{}, v8f{}};

  // mean @ Wl^T   (element (m,n) = sum_k mean[m,k] * Wl[n,k])
  #pragma unroll
  for (int k = 0; k < D; k += 4) {
    const int kk = k + 2 * half;
    v2f bw; bw.x = Wl[n * D + kk]; bw.y = Wl[n * D + kk + 1];
    #pragma unroll
    for (int mt = 0; mt < 4; ++mt) {
      v2f a;
      a.x = Am[mt * 16 + r][kk] * invc[mt];
      a.y = Am[mt * 16 + r][kk + 1] * invc[mt];
      acc[mt] = __builtin_amdgcn_wmma_f32_16x16x4_f32(false, a, false, bw,
                                                      (short)0, acc[mt],
                                                      false, false);
    }
  }
  // x_dst @ Wr^T
  #pragma unroll
  for (int k = 0; k < D; k += 4) {
    const int kk = k + 2 * half;
    v2f bw; bw.x = Wr[n * D + kk]; bw.y = Wr[n * D + kk + 1];
    #pragma unroll
    for (int mt = 0; mt < 4; ++mt) {
      v2f a;
      a.x = Ar[mt * 16 + r][kk];
      a.y = Ar[mt * 16 + r][kk + 1];
      acc[mt] = __builtin_amdgcn_wmma_f32_16x16x4_f32(false, a, false, bw,
                                                      (short)0, acc[mt],
                                                      false, false);
    }
  }

  const float bn = bias[n];
  #pragma unroll
  for (int mt = 0; mt < 4; ++mt) {
    #pragma unroll
    for (int v = 0; v < 8; ++v) {
      int m = mt * 16 + v + 8 * half;  // C/D layout: VGPR v, lane-half -> row
      int gr = row0 + m;
      if (gr < n_dst) {
        long long o = (long long)gr * D + n;
        h[o] = h[o] + acc[mt][v] + bn;  // stream-ordered accumulate across convs
      }
    }
  }
}

extern "C" void kernel_launch(void* const* d_in, const int* in_sizes, int n_in,
                              void* d_out, int out_size, void* d_ws, size_t ws_size,
                              hipStream_t stream) {
  (void)n_in; (void)out_size; (void)ws_size;

  const float* x_show  = (const float*)d_in[0];
  const float* x_song  = (const float*)d_in[1];
  const float* x_venue = (const float*)d_in[2];
  const int* ss_src = (const int*)d_in[3];  const int* ss_dst = (const int*)d_in[4];
  const int* sv_src = (const int*)d_in[5];  const int* sv_dst = (const int*)d_in[6];
  const int* vs_src = (const int*)d_in[7];  const int* vs_dst = (const int*)d_in[8];
  const int* hc_src = (const int*)d_in[9];  const int* hc_dst = (const int*)d_in[10];
  auto P = [&](int i) { return (const float*)d_in[i]; };
  const int E = in_sizes[3];

  // Workspace layout (floats): agg(reused, max show) | cnt | h_show | h_song | h_venue
  float* ws = (float*)d_ws;
  float* agg     = ws;
  float* cnt     = agg + (size_t)N_SHOW * D;
  float* h_show  = cnt + N_SHOW;
  float* h_song  = h_show + (size_t)N_SHOW * D;
  float* h_venue = h_song + (size_t)N_SONG * D;

  float* out     = (float*)d_out;
  float* o_show  = out;
  float* o_song  = out + (size_t)N_SHOW * D;
  float* o_venue = o_song + (size_t)N_SONG * D;

  auto zero = [&](float* p, long long n) {
    zero_f32<<<2048, 256, 0, stream>>>(p, n);
  };

  const int scat_blocks = (int)(((long long)E * 32 + 255) / 256);

  auto conv = [&](const float* xsrc, const float* xdst, int n_dst,
                  const int* src, const int* dst,
                  const float* Wl, const float* b, const float* Wr, float* h) {
    zero(agg, (long long)n_dst * D);
    zero(cnt, n_dst);
    scatter_mean_accum<<<scat_blocks, 256, 0, stream>>>(xsrc, src, dst, agg, cnt, E);
    sage_wmma_gemm<<<(n_dst + MT - 1) / MT, 256, 0, stream>>>(agg, cnt, xdst, Wl, b, Wr, h, n_dst);
  };

  // ---------------- Layer 1 ----------------
  zero(h_show,  (long long)N_SHOW * D);
  zero(h_song,  (long long)N_SONG * D);
  zero(h_venue, (long long)N_VENUE * D);
  conv(x_song,  x_show,  N_SHOW,  ss_src, ss_dst, P(11), P(12), P(13), h_show);   // ss
  conv(x_venue, x_show,  N_SHOW,  vs_src, vs_dst, P(17), P(18), P(19), h_show);   // vs
  conv(x_show,  x_venue, N_VENUE, sv_src, sv_dst, P(14), P(15), P(16), h_venue);  // sv
  conv(x_show,  x_song,  N_SONG,  hc_src, hc_dst, P(20), P(21), P(22), h_song);   // hc

  relu_f32<<<2048, 256, 0, stream>>>(h_show,  (long long)N_SHOW * D);
  relu_f32<<<2048, 256, 0, stream>>>(h_song,  (long long)N_SONG * D);
  relu_f32<<<2048, 256, 0, stream>>>(h_venue, (long long)N_VENUE * D);

  // ---------------- Layer 2 (writes directly to d_out) ----------------
  zero(out, (long long)(N_SHOW + N_SONG + N_VENUE) * D);
  conv(h_song,  h_show,  N_SHOW,  ss_src, ss_dst, P(23), P(24), P(25), o_show);
  conv(h_venue, h_show,  N_SHOW,  vs_src, vs_dst, P(29), P(30), P(31), o_show);
  conv(h_show,  h_venue, N_VENUE, sv_src, sv_dst, P(26), P(27), P(28), o_venue);
  conv(h_show,  h_song,  N_SONG,  hc_src, hc_dst, P(32), P(33), P(34), o_song);
}